// MDL_RNN_mate_2336462209346
// MI455X (gfx1250) — compile-verified
//
#include <hip/hip_runtime.h>

// ---------------------------------------------------------------------------
// SNN low-rank recurrent scan for MI455X (gfx1250, wave32)
//   kernel 1 (prep): zero-pad Wout -> Wpad[16][4096], zero R tail rows,
//                    init u-buffers + grid-barrier state
//   kernel 2 (scan): persistent 16-WGP scan, one hidden unit per lane,
//                    L2 atomics + generation grid-barrier, 3 rotating u-bufs
//   kernel 3 (gemm): Y = R @ Wout^T via V_WMMA_F32_16X16X4_F32, branch-free
//                    inner loop (padded operands keep EXEC all-ones)
// ---------------------------------------------------------------------------

#define HID    4096
#define TSTEPS 1000
#define RROWS  1008          // padded rows of R (tile 62 reads rows 992..1007)
#define NINP   6
#define PRANK  16
#define NWG    16
#define NTHR   256
#define REDPAD 17            // LDS stride padding (bank-conflict-free)

#define DT_C      0.005f
#define BETA_C    0.9f
#define THR_C     0.5f
#define TAU_MIN_C 0.02f
#define TAU_STP_C 0.03f
#define TAU_R_C   0.002f

typedef float v2f __attribute__((ext_vector_type(2)));
typedef float v8f __attribute__((ext_vector_type(8)));

// prep work items: 16*HID (Wpad) + 8*HID (R tail rows 1000..1007) + 50 (ctrl)
#define PREP_N (16 * HID + 8 * HID + 50)

__global__ void MDL_prep_kernel(const float* __restrict__ Wout,
                                float* __restrict__ Wpad,   // [16][HID]
                                float* __restrict__ Rtail,  // R + 1000*HID
                                float* __restrict__ ubuf,
                                unsigned* __restrict__ cnt,
                                unsigned* __restrict__ gen) {
    const int gid = blockIdx.x * 256 + threadIdx.x;
    if (gid < 16 * HID) {
        Wpad[gid] = (gid < 2 * HID) ? Wout[gid] : 0.0f;
    } else if (gid < 16 * HID + 8 * HID) {
        Rtail[gid - 16 * HID] = 0.0f;
    } else if (gid < PREP_N) {
        const int j = gid - (16 * HID + 8 * HID);
        if (j < 48) ubuf[j] = 0.0f;       // 3 rotating u-buffers of 16 floats
        else if (j == 48) *cnt = 0u;
        else              *gen = 0u;      // j == 49
    }
}

__global__ __launch_bounds__(NTHR, 1)
void MDL_scan_kernel(const float* __restrict__ x,     // [T,6]
                     const float* __restrict__ noise, // [T,H]
                     const float* __restrict__ Win,   // [H,6]
                     const float* __restrict__ pin,   // [H,16]
                     const float* __restrict__ pout,  // [H,16]
                     const float* __restrict__ l,     // [16]
                     const float* __restrict__ stau,  // [H]
                     float* __restrict__ R,           // [RROWS,H] workspace
                     float* __restrict__ ubuf,        // [3][16]
                     unsigned* __restrict__ cnt,
                     unsigned* __restrict__ gen) {
    const int tid = threadIdx.x;
    const int h   = blockIdx.x * NTHR + tid;

    __shared__ float red [NTHR * REDPAD];  // padded: conflict-free scatter
    __shared__ float red2[16 * 16];

    // --- per-unit constants held in registers for the whole scan ---
    float win[NINP];
#pragma unroll
    for (int i = 0; i < NINP; ++i) win[i] = Win[h * NINP + i];

    float pinl[PRANK], poutr[PRANK];
#pragma unroll
    for (int p = 0; p < PRANK; ++p) {
        const float lp = l[p];
        pinl[p]  = pin [h * PRANK + p] * lp;   // fold l into pin
        poutr[p] = pout[h * PRANK + p];
    }

    const float sig     = stau[h];
    const float tau_d   = TAU_STP_C / (1.0f + __expf(-sig)) + TAU_MIN_C;
    const float decay_d = __expf(-DT_C / tau_d);
    const float s_gain  = DT_C / (tau_d * TAU_R_C);
    const float decay_r = 0.082084998623898800f;  // exp(-DT/TAU_R) = exp(-2.5)

    float mem = 0.0f, r = 0.0f, s = 0.0f;

    for (int t = 0; t < TSTEPS; ++t) {
        // ---- read u(t) = pout^T r_{t-1} from rotating buffer (device scope) ----
        float u[PRANK];
        const float* ub = ubuf + (t % 3) * 16;
#pragma unroll
        for (int p = 0; p < PRANK; ++p)
            u[p] = __hip_atomic_load(ub + p, __ATOMIC_RELAXED,
                                     __HIP_MEMORY_SCOPE_AGENT);

        float v = 0.0f;
#pragma unroll
        for (int p = 0; p < PRANK; ++p) v += pinl[p] * u[p];

        float Iin = v + noise[(size_t)t * HID + h] * 0.1f;
#pragma unroll
        for (int i = 0; i < NINP; ++i) Iin += win[i] * x[t * NINP + i];

        // snntorch Leaky, subtract reset computed from incoming mem
        const float reset = (mem > THR_C) ? THR_C : 0.0f;
        mem = BETA_C * mem + Iin - reset;
        const float spk_gain = (mem > THR_C) ? s_gain : 0.0f;
        s = s * decay_r + spk_gain;
        r = decay_d * r + DT_C * s;

        R[(size_t)t * HID + h] = r;

        // ---- workgroup reduction of pout[p]*r into 16 partials ----
#pragma unroll
        for (int p = 0; p < PRANK; ++p) red[tid * REDPAD + p] = poutr[p] * r;
        __syncthreads();
        {
            const int c = tid >> 4, p = tid & 15;
            float sum = 0.0f;
#pragma unroll
            for (int j = 0; j < 16; ++j) sum += red[(c * 16 + j) * REDPAD + p];
            red2[c * 16 + p] = sum;
        }
        __syncthreads();
        if (tid < 16) {  // wave 0: accumulate u(t+1) partial into next buffer
            float sum = 0.0f;
#pragma unroll
            for (int c2 = 0; c2 < 16; ++c2) sum += red2[c2 * 16 + tid];
            __hip_atomic_fetch_add(ubuf + ((t + 1) % 3) * 16 + tid, sum,
                                   __ATOMIC_RELAXED, __HIP_MEMORY_SCOPE_AGENT);
        }

        // ---- grid barrier (generation counter through L2) ----
        __syncthreads();
        if (tid == 0) {
            // wave-0 fence drains the u-accumulation atomics from lanes 0..15
            __builtin_amdgcn_fence(__ATOMIC_RELEASE, "agent");
            const unsigned old = __hip_atomic_fetch_add(
                cnt, 1u, __ATOMIC_ACQ_REL, __HIP_MEMORY_SCOPE_AGENT);
            if (old == NWG - 1) {
                __hip_atomic_store(cnt, 0u, __ATOMIC_RELAXED,
                                   __HIP_MEMORY_SCOPE_AGENT);
                __hip_atomic_fetch_add(gen, 1u, __ATOMIC_RELEASE,
                                       __HIP_MEMORY_SCOPE_AGENT);
            } else {
                while (__hip_atomic_load(gen, __ATOMIC_ACQUIRE,
                                         __HIP_MEMORY_SCOPE_AGENT) <= (unsigned)t)
                    __builtin_amdgcn_s_sleep(2);
            }
        }
        __syncthreads();

        // ---- zero buf[t%3]; next accumulated at iter t+2 (post-barrier t+1) ----
        if (blockIdx.x == 0 && tid < 16)
            __hip_atomic_store(ubuf + (t % 3) * 16 + tid, 0.0f,
                               __ATOMIC_RELAXED, __HIP_MEMORY_SCOPE_AGENT);
    }
}

// Y[1000,2] = R[1000,4096] x Wout^T[4096,2] via f32 WMMA 16x16x4.
// One wave per block: 16 time rows, N padded to 16 (2 valid). Operands are
// zero-padded in the workspace so the inner loop is branch-free (EXEC stays
// all-ones, as WMMA requires).
__global__ __launch_bounds__(32, 1)
void MDL_out_gemm_kernel(const float* __restrict__ R,     // [RROWS][HID]
                         const float* __restrict__ Wpad,  // [16][HID], rows 2..15 = 0
                         float* __restrict__ y) {         // [1000, 2]
    const int lane = threadIdx.x;       // 0..31
    const int hi   = lane >> 4;         // K half selector (A/B: K=2*hi, 2*hi+1)
    const int l16  = lane & 15;         // row (A) / col (B)
    const int t0   = blockIdx.x * 16;

    const float* ra = R    + (size_t)(t0 + l16) * HID + hi * 2;  // rows < RROWS
    const float* wb = Wpad + (size_t)l16 * HID + hi * 2;

    v8f acc = {};
    for (int k = 0; k < HID; k += 4) {
        const v2f a = *(const v2f*)(ra + k);   // A[M=l16][K=k+2hi .. +1]
        const v2f b = *(const v2f*)(wb + k);   // B[K=k+2hi .. +1][N=l16]
        acc = __builtin_amdgcn_wmma_f32_16x16x4_f32(
            /*neg_a=*/false, a, /*neg_b=*/false, b,
            /*c_mod=*/(short)0, acc, /*reuse_a=*/false, /*reuse_b=*/false);
    }

    // D layout: VGPR v -> M = v + 8*(lane>=16); N = lane&15
    if (l16 < 2) {
#pragma unroll
        for (int vv = 0; vv < 8; ++vv) {
            const int tr = t0 + vv + hi * 8;
            if (tr < TSTEPS) y[tr * 2 + l16] = acc[vv];
        }
    }
}

extern "C" void kernel_launch(void* const* d_in, const int* in_sizes, int n_in,
                              void* d_out, int out_size, void* d_ws, size_t ws_size,
                              hipStream_t stream) {
    const float* x     = (const float*)d_in[0];
    const float* noise = (const float*)d_in[1];
    const float* Win   = (const float*)d_in[2];
    const float* Wout  = (const float*)d_in[3];
    const float* pin   = (const float*)d_in[4];
    const float* pout  = (const float*)d_in[5];
    const float* l     = (const float*)d_in[6];
    const float* stau  = (const float*)d_in[7];
    float* y = (float*)d_out;

    // workspace layout
    char* ws = (char*)d_ws;
    float* R = (float*)ws;                                   // RROWS*HID f32
    size_t off = (size_t)RROWS * HID * sizeof(float);        // 16,515,072 B
    float* Wpad = (float*)(ws + off);                        // 16*HID f32
    off += (size_t)16 * HID * sizeof(float);                 // +262,144 B
    float*    ubuf = (float*)(ws + off);                     // 48 floats
    unsigned* cnt  = (unsigned*)(ws + off + 64 * sizeof(float));
    unsigned* gen  = cnt + 1;

    hipLaunchKernelGGL(MDL_prep_kernel, dim3((PREP_N + 255) / 256), dim3(256),
                       0, stream, Wout, Wpad, R + (size_t)TSTEPS * HID,
                       ubuf, cnt, gen);
    hipLaunchKernelGGL(MDL_scan_kernel, dim3(NWG), dim3(NTHR), 0, stream,
                       x, noise, Win, pin, pout, l, stau, R, ubuf, cnt, gen);
    hipLaunchKernelGGL(MDL_out_gemm_kernel, dim3((TSTEPS + 15) / 16), dim3(32),
                       0, stream, R, Wpad, y);
}